// ConditionedPNA_8555574853800
// MI455X (gfx1250) — compile-verified
//
#include <hip/hip_runtime.h>
#include <math.h>

typedef __attribute__((ext_vector_type(2))) float v2f;
typedef __attribute__((ext_vector_type(8))) float v8f;

#define DD 128          // D (feature dim), also Q
#define KSTRIDE 132     // padded LDS row stride (132 % 64 == 4 -> conflict-free)

// ---- order-preserving float <-> uint encoding for atomic max/min ----
__device__ __forceinline__ unsigned encf(float f) {
    unsigned u = __float_as_uint(f);
    return (u & 0x80000000u) ? ~u : (u | 0x80000000u);
}
__device__ __forceinline__ float decf(unsigned u) {
    return __uint_as_float((u & 0x80000000u) ? (u & 0x7FFFFFFFu) : ~u);
}

// ---------------- init workspace ----------------
__global__ void pna_init_kernel(float* __restrict__ sumb, float* __restrict__ sqb,
                                unsigned* __restrict__ mxe, unsigned* __restrict__ mne,
                                float* __restrict__ logsum, long long count) {
    long long i = (long long)blockIdx.x * blockDim.x + threadIdx.x;
    if (i < count) {
        sumb[i] = 0.0f;
        sqb[i]  = 0.0f;
        mxe[i]  = 0x007FFFFFu;   // enc(-inf)
        mne[i]  = 0xFF800000u;   // enc(+inf)
    }
    if (i == 0) logsum[0] = 0.0f;
}

// ---------------- rel = query @ Wr^T + br ----------------
__global__ void pna_rel_kernel(const float* __restrict__ query, const float* __restrict__ Wr,
                               const float* __restrict__ br, float* __restrict__ rel, int M) {
    __shared__ float qs[DD];
    if (threadIdx.x < DD) qs[threadIdx.x] = query[threadIdx.x];
    __syncthreads();
    int i = blockIdx.x * blockDim.x + threadIdx.x;
    if (i >= M) return;
    const float* w = Wr + (size_t)i * DD;
    float acc = br[i];
#pragma unroll 8
    for (int k = 0; k < DD; ++k) acc = fmaf(qs[k], w[k], acc);
    rel[i] = acc;
}

// ---------------- sum of log(degree+1) over nodes ----------------
__global__ void pna_logdeg_kernel(const float* __restrict__ degree_out,
                                  float* __restrict__ logsum, int N) {
    __shared__ float sm[256];
    int i = blockIdx.x * blockDim.x + threadIdx.x;
    float v = (i < N) ? logf(degree_out[i] + 1.0f) : 0.0f;
    sm[threadIdx.x] = v;
    __syncthreads();
    for (int s = 128; s > 0; s >>= 1) {
        if (threadIdx.x < s) sm[threadIdx.x] += sm[threadIdx.x + s];
        __syncthreads();
    }
    if (threadIdx.x == 0) atomicAdd(logsum, sm[0]);
}

// ---------------- edge scatter: 1 wave per edge, 4 dims per lane ----------------
__global__ void pna_edge_kernel(const int* __restrict__ edge_index, const int* __restrict__ edge_attr,
                                const float* __restrict__ nf, const float* __restrict__ rel,
                                float* __restrict__ sumb, float* __restrict__ sqb,
                                unsigned* __restrict__ mxe, unsigned* __restrict__ mne, int E) {
    long long gid = (long long)blockIdx.x * blockDim.x + threadIdx.x;
    int e = (int)(gid >> 5);
    int l = (int)(gid & 31);
    if (e >= E) return;
    int srcn = edge_index[e];
    int dstn = edge_index[E + e];
    int att  = edge_attr[e];
    const float4 s4 = ((const float4*)(nf  + (size_t)srcn * DD))[l];
    const float4 r4 = ((const float4*)(rel + (size_t)att  * DD))[l];
    float m0 = s4.x * r4.x, m1 = s4.y * r4.y, m2 = s4.z * r4.z, m3 = s4.w * r4.w;
    size_t base = (size_t)dstn * DD + (size_t)l * 4;
    atomicAdd(&sumb[base + 0], m0);       atomicAdd(&sqb[base + 0], m0 * m0);
    atomicAdd(&sumb[base + 1], m1);       atomicAdd(&sqb[base + 1], m1 * m1);
    atomicAdd(&sumb[base + 2], m2);       atomicAdd(&sqb[base + 2], m2 * m2);
    atomicAdd(&sumb[base + 3], m3);       atomicAdd(&sqb[base + 3], m3 * m3);
    atomicMax(&mxe[base + 0], encf(m0));  atomicMin(&mne[base + 0], encf(m0));
    atomicMax(&mxe[base + 1], encf(m1));  atomicMin(&mne[base + 1], encf(m1));
    atomicMax(&mxe[base + 2], encf(m2));  atomicMin(&mne[base + 2], encf(m2));
    atomicMax(&mxe[base + 3], encf(m3));  atomicMin(&mne[base + 3], encf(m3));
}

// ---------------- per-(node,dim) epilogue; planes transformed in place ----------------
__global__ void pna_feat_kernel(const float* __restrict__ boundary, const float* __restrict__ degree_out,
                                const float* __restrict__ logsum,
                                float* __restrict__ sumb,   // -> mean
                                float* __restrict__ sqb,    // -> std
                                unsigned* __restrict__ mxe, // -> max (as float)
                                unsigned* __restrict__ mne, // -> min (as float)
                                float* __restrict__ scales, int N) {
    long long gid = (long long)blockIdx.x * blockDim.x + threadIdx.x;
    if (gid >= (long long)N * DD) return;
    int n = (int)(gid >> 7);
    int d = (int)(gid & (DD - 1));
    float deg0 = degree_out[n];
    float deg  = deg0 + 1.0f;
    float b    = boundary[gid];
    float s    = sumb[gid];
    float q    = sqb[gid];
    float mean   = (s + b) / deg;
    float sqmean = (q + b * b) / deg;
    float mx = (deg0 > 0.0f) ? decf(mxe[gid]) : 0.0f;
    float mn = (deg0 > 0.0f) ? decf(mne[gid]) : 0.0f;
    mx = fmaxf(mx, b);
    mn = fminf(mn, b);
    float sd = sqrtf(fmaxf(sqmean - mean * mean, 1e-6f));
    sumb[gid] = mean;
    ((float*)mxe)[gid] = mx;
    ((float*)mne)[gid] = mn;
    sqb[gid]  = sd;
    if (d == 0) {
        float meansc = logsum[0] / (float)N;
        float sc = logf(deg) / (meansc + 1e-10f);
        scales[n * 3 + 0] = 1.0f;
        scales[n * 3 + 1] = sc;
        scales[n * 3 + 2] = 1.0f / fmaxf(sc, 0.01f);
    }
}

// ---- assemble one element of the virtual X row: [nf | interleaved feats x scales] ----
__device__ __forceinline__ float pna_getA(int n, int k,
                                          const float* __restrict__ nf,
                                          const float* __restrict__ meanp,
                                          const float* __restrict__ maxp,
                                          const float* __restrict__ minp,
                                          const float* __restrict__ stdp,
                                          const float* __restrict__ scales) {
    if (k < DD) return nf[(size_t)n * DD + k];
    int kk = k - DD;
    int t = kk % 3;          // scale selector: 1, sc, 1/clip(sc)
    int f = kk / 3;          // feature index (d*4 + s)
    int s = f & 3;           // 0=mean 1=max 2=min 3=std
    int d = f >> 2;
    const float* plane = (s == 0) ? meanp : (s == 1) ? maxp : (s == 2) ? minp : stdp;
    return plane[(size_t)n * DD + d] * scales[n * 3 + t];
}

// ---------------- fused X-assembly + GEMM + bias + ReLU, f32 WMMA ----------------
// out[N,128] = relu( X[N,1664] @ Wlin[128,1664]^T + blin )
// block = 256 (8 waves): block owns 16 rows (m-tile); wave w owns cols [16w,16w+16)
__global__ void pna_gemm_kernel(const float* __restrict__ nf,
                                const float* __restrict__ meanp, const float* __restrict__ maxp,
                                const float* __restrict__ minp,  const float* __restrict__ stdp,
                                const float* __restrict__ scales,
                                const float* __restrict__ Wlin,  const float* __restrict__ blin,
                                float* __restrict__ out, int KT) {
    __shared__ float As[16 * KSTRIDE];
    const int m0   = blockIdx.x * 16;
    const int wave = threadIdx.x >> 5;
    const int lane = threadIdx.x & 31;
    const int n0   = wave * 16;
    const int ml    = lane & 15;            // row (A) / col (B,D) within tile
    const int khalf = (lane >> 4) * 2;      // K sub-offset: lanes 0-15 -> K0/K1, 16-31 -> K2/K3

    v8f acc = {};
    for (int kc = 0; kc < KT; kc += DD) {
        // cooperative assembly of 16 x 128 chunk of X into LDS
        for (int i = threadIdx.x; i < 16 * DD; i += 256) {
            int mm = i >> 7;
            int kk = i & (DD - 1);
            As[mm * KSTRIDE + kk] = pna_getA(m0 + mm, kc + kk, nf, meanp, maxp, minp, stdp, scales);
        }
        __syncthreads();

        const float* arow = &As[ml * KSTRIDE + khalf];
        const float* wrow = Wlin + (size_t)(n0 + ml) * KT + kc + khalf;
#pragma unroll 8
        for (int k = 0; k < DD; k += 4) {
            v2f a = { arow[k], arow[k + 1] };   // A[ml, kc+k+khalf .. +1]
            v2f b = { wrow[k], wrow[k + 1] };   // B[k'+khalf.., n0+ml] = Wlin[n0+ml, ...]
            acc = __builtin_amdgcn_wmma_f32_16x16x4_f32(
                false, a, false, b, (short)0, acc, false, false);
        }
        __syncthreads();
    }

    // D layout: VGPR r, lanes 0-15 -> M=r, lanes 16-31 -> M=r+8; N = lane&15
    const int moff = (lane >> 4) * 8;
    const int col  = n0 + ml;
    const float bias = blin[col];
#pragma unroll
    for (int r = 0; r < 8; ++r) {
        int row = m0 + r + moff;
        float v = acc[r] + bias;
        out[(size_t)row * DD + col] = fmaxf(v, 0.0f);
    }
}

extern "C" void kernel_launch(void* const* d_in, const int* in_sizes, int n_in,
                              void* d_out, int out_size, void* d_ws, size_t ws_size,
                              hipStream_t stream) {
    const float* nf       = (const float*)d_in[0];
    const float* query    = (const float*)d_in[1];
    const float* boundary = (const float*)d_in[2];
    const float* degree   = (const float*)d_in[3];
    const float* Wr       = (const float*)d_in[4];
    const float* br       = (const float*)d_in[5];
    const float* Wlin     = (const float*)d_in[6];
    const float* blin     = (const float*)d_in[7];
    const int*   edge_index = (const int*)d_in[8];
    const int*   edge_attr  = (const int*)d_in[9];

    const int N    = in_sizes[0] / DD;     // 50000
    const int E    = in_sizes[9];          // 600000
    const int M2RD = in_sizes[5];          // 2*R*D = 60672
    const int KT   = in_sizes[6] / DD;     // 13*D = 1664

    // workspace carve-up (256B aligned)
    auto align256 = [](size_t x) { return (x + 255) & ~(size_t)255; };
    char* p = (char*)d_ws;
    float*    rel    = (float*)p;    p += align256((size_t)M2RD * 4);
    float*    sumb   = (float*)p;    p += align256((size_t)N * DD * 4);
    float*    sqb    = (float*)p;    p += align256((size_t)N * DD * 4);
    unsigned* mxe    = (unsigned*)p; p += align256((size_t)N * DD * 4);
    unsigned* mne    = (unsigned*)p; p += align256((size_t)N * DD * 4);
    float*    scales = (float*)p;    p += align256((size_t)N * 3 * 4);
    float*    logsum = (float*)p;    p += align256(4);
    (void)ws_size; (void)n_in; (void)out_size;

    const long long nd = (long long)N * DD;

    // 1) init accumulator planes + scalar
    {
        long long blocks = (nd + 255) / 256;
        pna_init_kernel<<<(int)blocks, 256, 0, stream>>>(sumb, sqb, mxe, mne, logsum, nd);
    }
    // 2) relation features
    pna_rel_kernel<<<(M2RD + 255) / 256, 256, 0, stream>>>(query, Wr, br, rel, M2RD);
    // 3) sum of log(deg)
    pna_logdeg_kernel<<<(N + 255) / 256, 256, 0, stream>>>(degree, logsum, N);
    // 4) edge scatter (1 wave per edge)
    {
        long long threads = (long long)E * 32;
        pna_edge_kernel<<<(int)((threads + 255) / 256), 256, 0, stream>>>(
            edge_index, edge_attr, nf, rel, sumb, sqb, mxe, mne, E);
    }
    // 5) per-node epilogue (planes -> mean/std/max/min in place) + scales
    pna_feat_kernel<<<(int)((nd + 255) / 256), 256, 0, stream>>>(
        boundary, degree, logsum, sumb, sqb, mxe, mne, scales, N);
    // 6) fused assemble + WMMA GEMM + bias + ReLU
    pna_gemm_kernel<<<N / 16, 256, 0, stream>>>(
        nf, sumb, (const float*)mxe, (const float*)mne, sqb, scales,
        Wlin, blin, (float*)d_out, KT);
}